// LearnableDiffusionContrastiveLoss_27504970563864
// MI455X (gfx1250) — compile-verified
//
#include <hip/hip_runtime.h>
#include <hip/hip_bf16.h>
#include <math.h>

// ---------------------------------------------------------------------------
// LearnableDiffusionContrastiveLoss for MI455X (gfx1250, wave32, WMMA).
//
// Roofline: dm is 256 MiB fp32; the loss GEMM is 34.4 GFLOP -> memory bound
// (~3 TFLOP/s needed over the 11.5us dm read at 23.3 TB/s). Design:
//   * read dm exactly twice (rowsum pass + GEMM pass), non-temporal so the
//     256 MiB stream does not evict z (8 MiB, L2-resident) from the 192MB L2,
//   * native fp32 WMMA (V_WMMA_F32_16X16X4_F32), no precision downgrade,
//   * B staged K-contiguous in LDS so every WMMA operand is one aligned
//     ds_load_b64 (no v_dual_mov repacking),
//   * global->LDS staging via GLOBAL_LOAD_ASYNC_TO_LDS (ASYNCcnt) when the
//     toolchain exposes the builtins; clean synchronous fallback otherwise.
// ---------------------------------------------------------------------------

#define N_NODES 8192
#define DF      256
#define NEG     10
#define INV_TEMP 5.0f     // 1/0.2
#define KC      32        // K chunk staged in LDS
#define ASTR    36        // LDS row stride (floats): 16B-aligned rows, and
                          // 36*L mod 64 -> conflict-free across 16 lanes

typedef __attribute__((ext_vector_type(2))) float v2f;
typedef __attribute__((ext_vector_type(4))) float v4f;
typedef __attribute__((ext_vector_type(8))) float v8f;
typedef int i4v __attribute__((vector_size(16)));   // matches builtin proto

#if defined(__has_builtin)
#if __has_builtin(__builtin_amdgcn_global_load_async_to_lds_b32) && \
    __has_builtin(__builtin_amdgcn_global_load_async_to_lds_b128) && \
    __has_builtin(__builtin_amdgcn_s_wait_asynccnt)
#define USE_ASYNC_LDS 1
#endif
#endif

// Typed address-space casts (generic -> global AS1 / LDS AS3).
#define G_PTR(T, p) ((__attribute__((address_space(1))) T*)(p))
#define L_PTR(T, p) ((__attribute__((address_space(3))) T*)(p))

// ---------------------------------------------------------------------------
// Kernel 1: per-row sum of dm + 10-step scalar normalization recurrence.
// dm_k = dm0 / C_k with C_0 = 1, C_{k+1} = C_k * rowsum(dm_k) = C_k * (S/C_k).
// One wave per row; non-temporal b128 loads (dm is a one-shot stream).
// ---------------------------------------------------------------------------
__global__ void rowsum_k(const float* __restrict__ dm,
                         float* __restrict__ invscale,
                         float* __restrict__ lossacc) {
    if (blockIdx.x == 0 && threadIdx.x == 0) *lossacc = 0.0f;
    const int row  = (blockIdx.x << 3) + (threadIdx.x >> 5);
    const int lane = threadIdx.x & 31;
    const v4f* p = (const v4f*)(dm + (size_t)row * N_NODES);
    float s = 0.0f;
    for (int i = lane; i < N_NODES / 4; i += 32) {
        v4f v = __builtin_nontemporal_load(p + i);
        s += (v.x + v.y) + (v.z + v.w);
    }
    #pragma unroll
    for (int off = 16; off > 0; off >>= 1) s += __shfl_xor(s, off, 32);
    if (lane == 0) {
        float S = s, C = 1.0f;
        #pragma unroll
        for (int k = 0; k < 10; ++k) { float r = S / C; C *= r; }
        invscale[row] = 1.0f / C;
    }
}

// ---------------------------------------------------------------------------
// Kernel 2: gather + L2-normalize the NEG negative rows (reference:
// negative = z[neg_idx]).  One wave per negative row.
// ---------------------------------------------------------------------------
__global__ void negnorm_k(const float* __restrict__ z,
                          const int* __restrict__ neg_idx,
                          float* __restrict__ nneg) {
    const int w = threadIdx.x >> 5;
    const int lane = threadIdx.x & 31;
    if (w >= NEG) return;
    const int src = neg_idx[w];
    const float* zr = z + (size_t)src * DF;
    float v[8];
    float s = 0.0f;
    #pragma unroll
    for (int i = 0; i < 8; ++i) { v[i] = zr[lane + 32 * i]; s += v[i] * v[i]; }
    #pragma unroll
    for (int off = 16; off > 0; off >>= 1) s += __shfl_xor(s, off, 32);
    const float inv = 1.0f / fmaxf(sqrtf(s), 1e-12f);
    #pragma unroll
    for (int i = 0; i < 8; ++i) nneg[w * DF + lane + 32 * i] = v[i] * inv;
}

// ---------------------------------------------------------------------------
// Kernel 3: fused  positive = diag(invscale) * (dm @ z)  +  InfoNCE loss.
// Block: 256 threads (8 waves), output tile = 32 rows x 256 cols (full DF).
// Wave w: rows 16*(w>>2)..+16, cols 64*(w&3)..+64 -> 4 WMMA C tiles (v8f).
// A staged row-major [32][ASTR]; B staged transposed [256][ASTR] so both
// WMMA operands are single aligned b64 LDS reads.  Epilogue reuses BT_s for
// the scaled positive tile and computes the per-row contrastive loss.
// ---------------------------------------------------------------------------
__global__ void __launch_bounds__(256)
diffuse_loss_k(const float* __restrict__ dm,
               const float* __restrict__ z,
               const float* __restrict__ invscale,
               const float* __restrict__ nneg,
               float* __restrict__ lossacc) {
    __shared__ float A_s[32][ASTR];    //  4.6 KB: dm tile, row-major
    __shared__ float BT_s[DF][ASTR];   // 36.9 KB: z tile, K-contiguous

    const int tid    = threadIdx.x;
    const int wave   = tid >> 5;
    const int lane   = tid & 31;
    const int m16    = lane & 15;          // row/col within 16-wide tile
    const int koff   = (lane >> 4) << 1;   // K sub-offset: lanes16-31 -> K+2
    const int rb16   = (wave >> 2) << 4;   // wave's local row base (0 or 16)
    const int cbase  = (wave & 3) << 6;    // wave's local col base (0..192)
    const int rowBase = blockIdx.x << 5;   // 32 rows per block

    const v8f vzero = {0.f, 0.f, 0.f, 0.f, 0.f, 0.f, 0.f, 0.f};
    v8f acc[4];
    #pragma unroll
    for (int t = 0; t < 4; ++t) acc[t] = vzero;

    // Per-thread A staging coordinates (one b128 per thread per chunk).
    const int ar  = tid >> 3;          // 0..31 : A row
    const int ac4 = (tid & 7) << 2;    // 0..28 : A col (float4)

    for (int kb = 0; kb < N_NODES; kb += KC) {
        const float* aptr = dm + (size_t)(rowBase + ar) * N_NODES + kb + ac4;
#ifdef USE_ASYNC_LDS
        // ---- async global->LDS staging, tracked by ASYNCcnt --------------
        __builtin_amdgcn_global_load_async_to_lds_b128(
            G_PTR(i4v, aptr), L_PTR(i4v, &A_s[ar][ac4]), 0, 0);
        #pragma unroll
        for (int i = 0; i < KC; ++i) {   // coalesced across lanes per i
            __builtin_amdgcn_global_load_async_to_lds_b32(
                G_PTR(int, z + (size_t)(kb + i) * DF + tid),
                L_PTR(int, &BT_s[tid][i]), 0, 0);
        }
#else
        // ---- synchronous fallback ----------------------------------------
        const v4f av = __builtin_nontemporal_load((const v4f*)aptr);
        *(v4f*)(&A_s[ar][ac4]) = av;
        #pragma unroll
        for (int i4 = 0; i4 < KC; i4 += 4) {   // thread tid owns column tid
            v4f bv;
            bv.x = z[(size_t)(kb + i4 + 0) * DF + tid];
            bv.y = z[(size_t)(kb + i4 + 1) * DF + tid];
            bv.z = z[(size_t)(kb + i4 + 2) * DF + tid];
            bv.w = z[(size_t)(kb + i4 + 3) * DF + tid];
            *(v4f*)(&BT_s[tid][i4]) = bv;      // own-row store, conflict-free
        }
#endif
        // ---- prefetch next dm chunk (global_prefetch_b8) -----------------
        if (kb + KC < N_NODES) {
            __builtin_prefetch(aptr + KC, 0, 3);
        }
#ifdef USE_ASYNC_LDS
        __builtin_amdgcn_s_wait_asynccnt(0);
#endif
        __syncthreads();

        // ---- 8 WMMA K-steps of 4, 4 column tiles each --------------------
        #pragma unroll
        for (int kk = 0; kk < KC; kk += 4) {
            const v2f a = *(const v2f*)&A_s[rb16 + m16][kk + koff];
            #pragma unroll
            for (int t = 0; t < 4; ++t) {
                const int col = cbase + (t << 4) + m16;
                const v2f b = *(const v2f*)&BT_s[col][kk + koff];
                acc[t] = __builtin_amdgcn_wmma_f32_16x16x4_f32(
                    false, a, false, b, (short)0, acc[t], false, false);
            }
        }
        __syncthreads();
    }

    // ---- epilogue: scale by invscale, park positive tile in BT_s ----------
    float* pos = &BT_s[0][0];            // reuse as [32][256] row-major
    float sc[8];
    #pragma unroll
    for (int v = 0; v < 8; ++v)
        sc[v] = invscale[rowBase + rb16 + v + ((lane >> 4) << 3)];
    #pragma unroll
    for (int t = 0; t < 4; ++t) {
        #pragma unroll
        for (int v = 0; v < 8; ++v) {
            const int lr = rb16 + v + ((lane >> 4) << 3);     // local row 0..31
            pos[lr * DF + cbase + (t << 4) + m16] = acc[t][v] * sc[v];
        }
    }
    __syncthreads();

    // ---- per-row loss: 8 threads per row, 32 cols each --------------------
    const int row = tid >> 3;          // 0..31
    const int sub = tid & 7;           // 0..7 (8-lane group within wave)
    const int grow = rowBase + row;
    const float* zr = z + (size_t)grow * DF;

    float aa = 0.f, ap = 0.f, pp = 0.f;
    float nd[NEG];
    #pragma unroll
    for (int k = 0; k < NEG; ++k) nd[k] = 0.f;

    for (int i = 0; i < 32; ++i) {
        const int c = (sub << 5) + i;
        const float avl = zr[c];
        const float pvl = pos[row * DF + c];
        aa += avl * avl;
        ap += avl * pvl;
        pp += pvl * pvl;
        #pragma unroll
        for (int k = 0; k < NEG; ++k) nd[k] += avl * nneg[k * DF + c];
    }
    #pragma unroll
    for (int off = 4; off > 0; off >>= 1) {
        aa += __shfl_xor(aa, off, 8);
        ap += __shfl_xor(ap, off, 8);
        pp += __shfl_xor(pp, off, 8);
        #pragma unroll
        for (int k = 0; k < NEG; ++k) nd[k] += __shfl_xor(nd[k], off, 8);
    }
    if (sub == 0) {
        const float inva = 1.0f / fmaxf(sqrtf(aa), 1e-12f);
        const float invp = 1.0f / fmaxf(sqrtf(pp), 1e-12f);
        const float pos_sim = ap * inva * invp * INV_TEMP;
        float negsum = 0.f;
        #pragma unroll
        for (int k = 0; k < NEG; ++k)
            negsum += __expf(nd[k] * inva * INV_TEMP);
        // -log(e^p / (e^p + negsum)) = log(e^p + negsum) - p
        const float li = __logf(__expf(pos_sim) + negsum) - pos_sim;
        atomicAdd(lossacc, li);
    }
}

// ---------------------------------------------------------------------------
// Kernel 4: mean.
// ---------------------------------------------------------------------------
__global__ void finalize_k(const float* __restrict__ lossacc,
                           float* __restrict__ out) {
    if (threadIdx.x == 0) out[0] = lossacc[0] * (1.0f / (float)N_NODES);
}

// ---------------------------------------------------------------------------
extern "C" void kernel_launch(void* const* d_in, const int* in_sizes, int n_in,
                              void* d_out, int out_size, void* d_ws,
                              size_t ws_size, hipStream_t stream) {
    const float* z       = (const float*)d_in[0];   // [8192,256] f32
    const float* dm      = (const float*)d_in[1];   // [8192,8192] f32
    // d_in[2] edge_index, d_in[3] reliable_negatives: unused by the forward
    const int*   neg_idx = (const int*)d_in[4];     // [10] int32

    float* invscale = (float*)d_ws;                 // 8192 floats
    float* nneg     = invscale + N_NODES;           // 10*256 floats
    float* lossacc  = nneg + NEG * DF;              // 1 float

    rowsum_k<<<N_NODES / 8, 256, 0, stream>>>(dm, invscale, lossacc);
    negnorm_k<<<1, 320, 0, stream>>>(z, neg_idx, nneg);
    diffuse_loss_k<<<N_NODES / 32, 256, 0, stream>>>(dm, z, invscale, nneg,
                                                     lossacc);
    finalize_k<<<1, 64, 0, stream>>>(lossacc, (float*)d_out);
}